// ListFoldLossOptimized_16793322127950
// MI455X (gfx1250) — compile-verified
//
#include <hip/hip_runtime.h>
#include <math.h>

// ListFold loss for N = 8388608 float32 elements on MI455X (gfx1250, wave32).
// Pipeline: stable LSD radix sort (4x8-bit) of (key=-target, val=pred),
// exp + dual inclusive scan (3-kernel decoupled), then a pair-term reduction
// that accumulates through V_WMMA_F32_16X16X4_F32 (all-ones B => row-sum
// accumulate, layout-agnostic, f32 precision). All reductions are fixed-order
// trees -> bitwise deterministic across graph replays.
//
// Workspace layout (n = 8,388,608):
//   keyA[n] u32 | valA[n] u32 | keyB[n] u32 | valB[n] u32   (128 MB)
//   hist[256 * (n/4096)] u32                                 (2 MB)
//   part[n/4096] float2                                      (16 KB)
//   fpart[(n/2)/4096] float                                  (4 KB)
// After the sort, keyA/keyB are reused as the cumsum arrays Ip/Im.

typedef unsigned int u32;
typedef __attribute__((ext_vector_type(2))) float v2f;
typedef __attribute__((ext_vector_type(8))) float v8f;

#define SORT_BLOCK 512
#define SORT_WAVES (SORT_BLOCK / 32)       // 16
#define SORT_ITEMS 8
#define SORT_TILE  (SORT_BLOCK * SORT_ITEMS) // 4096

#define RED_BLOCK 256
#define RED_SEG   4096

#define SW_BLOCK 256
#define SW_ITEMS 16
#define SW_SEG   4096

#define F_BLOCK 256
#define F_WAVES 8
#define F_WITER 8   // per wave: 8 wmma iters * 64 terms = 512; per block = 4096 terms

// ---------------------------------------------------------------------------
// 0) key transform: descending-by-target == ascending radix on sortable(-target)
__global__ void k_transform(const float* __restrict__ target,
                            const float* __restrict__ pred,
                            u32* __restrict__ keyOut, u32* __restrict__ valOut,
                            int n) {
  int i = blockIdx.x * blockDim.x + threadIdx.x;
  if (i < n) {
    u32 u = __float_as_uint(-target[i]);
    u = (u & 0x80000000u) ? ~u : (u | 0x80000000u);
    keyOut[i] = u;
    valOut[i] = __float_as_uint(pred[i]);
  }
}

// ---------------------------------------------------------------------------
// 1) per-block 256-bucket histograms, bucket-major layout hist[d*NB + blk]
__global__ void k_hist(const u32* __restrict__ keys, u32* __restrict__ hist,
                       int shift, int nblocks) {
  __shared__ u32 h[256];
  int t = threadIdx.x;
  for (int d = t; d < 256; d += SORT_BLOCK) h[d] = 0;
  __syncthreads();
  size_t base = (size_t)blockIdx.x * SORT_TILE;
#pragma unroll
  for (int it = 0; it < SORT_ITEMS; ++it) {
    u32 k = keys[base + (size_t)it * SORT_BLOCK + t];
    atomicAdd(&h[(k >> shift) & 255u], 1u);
  }
  __syncthreads();
  for (int d = t; d < 256; d += SORT_BLOCK)
    hist[(size_t)d * nblocks + blockIdx.x] = h[d];
}

// ---------------------------------------------------------------------------
// 2) single-block exclusive scan over the 256*NB histogram (chunked, carried)
__global__ void k_scan_hist(u32* __restrict__ hist, int nh) {
  __shared__ u32 wsum[32];
  __shared__ u32 carry_s;
  int t = threadIdx.x, lane = t & 31, wid = t >> 5;
  u32 carry = 0;
  for (int base = 0; base < nh; base += 1024) {
    u32 x = hist[base + t];
    u32 incl = x;
#pragma unroll
    for (int d = 1; d < 32; d <<= 1) {
      u32 y = __shfl_up(incl, d, 32);
      if (lane >= d) incl += y;
    }
    if (lane == 31) wsum[wid] = incl;
    __syncthreads();
    if (wid == 0) {
      u32 v = wsum[lane];
      u32 iv = v;
#pragma unroll
      for (int d = 1; d < 32; d <<= 1) {
        u32 y = __shfl_up(iv, d, 32);
        if (lane >= d) iv += y;
      }
      wsum[lane] = iv - v;  // exclusive wave offset
    }
    __syncthreads();
    u32 excl = carry + wsum[wid] + incl - x;
    hist[base + t] = excl;
    if (t == 1023) carry_s = excl + x;  // carry + chunk total
    __syncthreads();
    carry = carry_s;
    __syncthreads();
  }
}

// ---------------------------------------------------------------------------
// 3) stable scatter. Tile element order = (wave, item, lane); ranks built from
//    wave32 ballots + per-wave LDS counters only -> fully stable/deterministic.
__global__ void k_scatter(const u32* __restrict__ keysIn, const u32* __restrict__ valsIn,
                          u32* __restrict__ keysOut, u32* __restrict__ valsOut,
                          const u32* __restrict__ hist, int shift, int nblocks) {
  __shared__ u32 wcnt[SORT_WAVES][256];
  __shared__ u32 gb[256];
  int t = threadIdx.x, lane = t & 31, w = t >> 5;
  for (int i = t; i < SORT_WAVES * 256; i += SORT_BLOCK) (&wcnt[0][0])[i] = 0;
  for (int d = t; d < 256; d += SORT_BLOCK)
    gb[d] = hist[(size_t)d * nblocks + blockIdx.x];
  __syncthreads();

  size_t wbase = (size_t)blockIdx.x * SORT_TILE + (size_t)w * (32 * SORT_ITEMS);
  u32 k[SORT_ITEMS], v[SORT_ITEMS], lr[SORT_ITEMS], dg[SORT_ITEMS];
#pragma unroll
  for (int it = 0; it < SORT_ITEMS; ++it) {
    size_t idx = wbase + (size_t)it * 32 + lane;
    u32 kk = keysIn[idx];
    k[it] = kk;
    v[it] = valsIn[idx];
    u32 d = (kk >> shift) & 255u;
    dg[it] = d;
    // peer mask: lanes in this wave holding the same digit
    u32 peers = 0xFFFFFFFFu;
#pragma unroll
    for (int b = 0; b < 8; ++b) {
      int bit = (int)((d >> b) & 1u);
      u32 vote = (u32)__ballot(bit);
      peers &= bit ? vote : ~vote;
    }
    u32 rank   = __popc(peers & ((1u << lane) - 1u));
    u32 leader = (u32)(__ffs(peers) - 1);
    u32 cnt    = __popc(peers);
    u32 base_d = 0;
    if ((u32)lane == leader) base_d = atomicAdd(&wcnt[w][d], cnt);
    base_d = __shfl(base_d, (int)leader, 32);
    lr[it] = base_d + rank;
  }
  __syncthreads();
  // exclusive prefix over waves, per digit (thread d walks 16 waves)
  if (t < 256) {
    u32 run = 0;
#pragma unroll
    for (int ww = 0; ww < SORT_WAVES; ++ww) {
      u32 c = wcnt[ww][t];
      wcnt[ww][t] = run;
      run += c;
    }
  }
  __syncthreads();
#pragma unroll
  for (int it = 0; it < SORT_ITEMS; ++it) {
    u32 d = dg[it];
    size_t pos = (size_t)gb[d] + wcnt[w][d] + lr[it];
    keysOut[pos] = k[it];
    valsOut[pos] = v[it];
  }
}

// ---------------------------------------------------------------------------
// 4) per-block sums of exp(+sp), exp(-sp)
__global__ void k_exp_partials(const float* __restrict__ sp, float2* __restrict__ part) {
  int t = threadIdx.x, lane = t & 31, w = t >> 5;
  size_t base = (size_t)blockIdx.x * RED_SEG;
  float ap = 0.f, am = 0.f;
  for (int i = t; i < RED_SEG; i += RED_BLOCK) {
    float x = sp[base + i];
    ap += expf(x);
    am += expf(-x);
  }
#pragma unroll
  for (int off = 16; off >= 1; off >>= 1) {
    ap += __shfl_down(ap, off, 32);
    am += __shfl_down(am, off, 32);
  }
  __shared__ float wp[8], wm[8];
  if (lane == 0) { wp[w] = ap; wm[w] = am; }
  __syncthreads();
  if (t == 0) {
    float sp_ = 0.f, sm_ = 0.f;
    for (int i = 0; i < 8; ++i) { sp_ += wp[i]; sm_ += wm[i]; }
    part[blockIdx.x] = make_float2(sp_, sm_);
  }
}

// ---------------------------------------------------------------------------
// 5) single-block exclusive scan of the block partial pairs (chunked, carried)
__global__ void k_scan_partials(float2* __restrict__ part, int ns) {
  __shared__ float wsp[32], wsm[32];
  __shared__ float cps, cms;
  int t = threadIdx.x, lane = t & 31, wid = t >> 5;
  float carryP = 0.f, carryM = 0.f;
  for (int base = 0; base < ns; base += 1024) {
    float2 x = part[base + t];
    float ip = x.x, im = x.y;
#pragma unroll
    for (int d = 1; d < 32; d <<= 1) {
      float yp = __shfl_up(ip, d, 32), ym = __shfl_up(im, d, 32);
      if (lane >= d) { ip += yp; im += ym; }
    }
    if (lane == 31) { wsp[wid] = ip; wsm[wid] = im; }
    __syncthreads();
    if (wid == 0) {
      float vp = wsp[lane], vm = wsm[lane];
      float ivp = vp, ivm = vm;
#pragma unroll
      for (int d = 1; d < 32; d <<= 1) {
        float yp = __shfl_up(ivp, d, 32), ym = __shfl_up(ivm, d, 32);
        if (lane >= d) { ivp += yp; ivm += ym; }
      }
      wsp[lane] = ivp - vp;
      wsm[lane] = ivm - vm;
    }
    __syncthreads();
    float ep = carryP + wsp[wid] + ip - x.x;
    float em = carryM + wsm[wid] + im - x.y;
    part[base + t] = make_float2(ep, em);
    if (t == 1023) { cps = ep + x.x; cms = em + x.y; }
    __syncthreads();
    carryP = cps;
    carryM = cms;
    __syncthreads();
  }
}

// ---------------------------------------------------------------------------
// 6) write inclusive cumsums Ip/Im (recompute exp; block offset from part[])
__global__ void k_scan_write(const float* __restrict__ sp, const float2* __restrict__ offs,
                             float* __restrict__ Ip, float* __restrict__ Im) {
  int t = threadIdx.x, lane = t & 31, w = t >> 5;
  size_t base = (size_t)blockIdx.x * SW_SEG + (size_t)t * SW_ITEMS;
  float p[SW_ITEMS], m[SW_ITEMS];
  float ap = 0.f, am = 0.f;
#pragma unroll
  for (int it = 0; it < SW_ITEMS; ++it) {
    float x = sp[base + it];
    ap += expf(x);
    am += expf(-x);
    p[it] = ap;
    m[it] = am;
  }
  // exclusive prefix of per-thread totals over the 256-thread block
  float ipv = ap, imv = am;
#pragma unroll
  for (int d = 1; d < 32; d <<= 1) {
    float yp = __shfl_up(ipv, d, 32), ym = __shfl_up(imv, d, 32);
    if (lane >= d) { ipv += yp; imv += ym; }
  }
  __shared__ float wp[8], wm[8], owp[8], owm[8];
  if (lane == 31) { wp[w] = ipv; wm[w] = imv; }
  __syncthreads();
  if (t == 0) {
    float rp = 0.f, rm = 0.f;
    for (int i = 0; i < 8; ++i) { owp[i] = rp; owm[i] = rm; rp += wp[i]; rm += wm[i]; }
  }
  __syncthreads();
  float2 off = offs[blockIdx.x];
  float basP = off.x + owp[w] + (ipv - ap);
  float basM = off.y + owm[w] + (imv - am);
#pragma unroll
  for (int it = 0; it < SW_ITEMS; ++it) {
    Ip[base + it] = p[it] + basP;
    Im[base + it] = m[it] + basM;
  }
}

// ---------------------------------------------------------------------------
// 7) pair terms + WMMA f32 accumulate. den_i = S+(i)*S-(i) - (n-2i);
//    term_i = log(den_i) - (sp[i] - sp[n-1-i]). A-fragment holds 64 terms,
//    B = all-ones -> D row-sums accumulate; sum(all C)/16 = exact total.
__device__ __forceinline__ float pair_term(long long i, long long n,
                                           const float* __restrict__ sp,
                                           const float* __restrict__ Ip,
                                           const float* __restrict__ Im) {
  float spi = sp[i];
  float spn = sp[n - 1 - i];
  float ph = Ip[n - 1 - i];   // cp[n-i] == inclusive[n-1-i]
  float mh = Im[n - 1 - i];
  float pl = 0.f, ml = 0.f;
  if (i > 0) { pl = Ip[i - 1]; ml = Im[i - 1]; }
  float s_plus = ph - pl, s_minus = mh - ml;
  float L = (float)(n - 2 * i);
  float den = fmaf(s_plus, s_minus, -L);
  return logf(den) - (spi - spn);
}

__global__ void k_pair_reduce(const float* __restrict__ sp, const float* __restrict__ Ip,
                              const float* __restrict__ Im, float* __restrict__ fpart,
                              long long n) {
  int t = threadIdx.x, lane = t & 31, w = t >> 5;
  int lo = lane & 15, hi = lane >> 4;
  long long wbase = ((long long)blockIdx.x * F_WAVES + w) * (long long)(F_WITER * 64);
  v8f c = {0.f, 0.f, 0.f, 0.f, 0.f, 0.f, 0.f, 0.f};
  v2f b;
  b[0] = 1.0f;
  b[1] = 1.0f;
  for (int j = 0; j < F_WITER; ++j) {
    long long base = wbase + (long long)j * 64;
    long long ia = base + (long long)(lo * 4 + 2 * hi);
    v2f a;
    a[0] = pair_term(ia, n, sp, Ip, Im);
    a[1] = pair_term(ia + 1, n, sp, Ip, Im);
    // D = A(16x4) x ones(4x16) + C : accumulates row-sums of the 64 terms
    c = __builtin_amdgcn_wmma_f32_16x16x4_f32(false, a, false, b, (short)0, c,
                                              false, false);
  }
  float s = c[0] + c[1] + c[2] + c[3] + c[4] + c[5] + c[6] + c[7];
#pragma unroll
  for (int off = 16; off >= 1; off >>= 1) s += __shfl_down(s, off, 32);
  __shared__ float ws_[F_WAVES];
  if (lane == 0) ws_[w] = s * 0.0625f;  // sum(all C) == 16 * wave total
  __syncthreads();
  if (t == 0) {
    float tot = 0.f;
    for (int i = 0; i < F_WAVES; ++i) tot += ws_[i];
    fpart[blockIdx.x] = tot;
  }
}

// ---------------------------------------------------------------------------
// 8) deterministic final tree reduce of block partials -> d_out[0]
__global__ void k_final(const float* __restrict__ fpart, float* __restrict__ out, int nf) {
  __shared__ float s[1024];
  int t = threadIdx.x;
  s[t] = (t < nf) ? fpart[t] : 0.f;
  __syncthreads();
  for (int st = 512; st >= 1; st >>= 1) {
    if (t < st) s[t] += s[t + st];
    __syncthreads();
  }
  if (t == 0) out[0] = s[0];
}

// ---------------------------------------------------------------------------
extern "C" void kernel_launch(void* const* d_in, const int* in_sizes, int n_in,
                              void* d_out, int out_size, void* d_ws, size_t ws_size,
                              hipStream_t stream) {
  const float* pred = (const float*)d_in[0];
  const float* target = (const float*)d_in[1];
  int n = in_sizes[0];  // 8388608 (power of two; all tiles divide exactly)

  char* ws = (char*)d_ws;
  size_t nb4 = (size_t)n * 4;
  u32* keyA = (u32*)(ws + 0 * nb4);
  u32* valA = (u32*)(ws + 1 * nb4);
  u32* keyB = (u32*)(ws + 2 * nb4);
  u32* valB = (u32*)(ws + 3 * nb4);
  int NB = n / SORT_TILE;  // 2048
  u32* hist = (u32*)(ws + 4 * nb4);
  size_t histBytes = (size_t)256 * NB * sizeof(u32);
  int NS = n / RED_SEG;  // 2048
  float2* part = (float2*)(ws + 4 * nb4 + histBytes);
  float* fpart = (float*)((char*)part + (size_t)NS * sizeof(float2));
  long long npairs = (long long)n / 2;
  int NF = (int)(npairs / ((long long)F_BLOCK / 32 * F_WITER * 64));  // 1024

  k_transform<<<n / 256, 256, 0, stream>>>(target, pred, keyA, valA, n);

  u32 *ki = keyA, *vi = valA, *ko = keyB, *vo = valB;
  for (int pass = 0; pass < 4; ++pass) {
    int shift = pass * 8;
    k_hist<<<NB, SORT_BLOCK, 0, stream>>>(ki, hist, shift, NB);
    k_scan_hist<<<1, 1024, 0, stream>>>(hist, 256 * NB);
    k_scatter<<<NB, SORT_BLOCK, 0, stream>>>(ki, vi, ko, vo, hist, shift, NB);
    u32* tk = ki; ki = ko; ko = tk;
    u32* tv = vi; vi = vo; vo = tv;
  }
  // after 4 passes sorted data is back in keyA/valA (ki/vi)
  float* sp = (float*)vi;   // sorted pred
  float* Ip = (float*)ki;   // reuse key buffer for cumsum(exp(+sp))
  float* Im = (float*)ko;   // reuse other key buffer for cumsum(exp(-sp))

  k_exp_partials<<<NS, RED_BLOCK, 0, stream>>>(sp, part);
  k_scan_partials<<<1, 1024, 0, stream>>>(part, NS);
  k_scan_write<<<NS, SW_BLOCK, 0, stream>>>(sp, part, Ip, Im);
  k_pair_reduce<<<NF, F_BLOCK, 0, stream>>>(sp, Ip, Im, fpart, (long long)n);
  k_final<<<1, 1024, 0, stream>>>(fpart, (float*)d_out, NF);
}